// FFT_CNN_40458591928810
// MI455X (gfx1250) — compile-verified
//
#include <hip/hip_runtime.h>
#include <math.h>

typedef __attribute__((ext_vector_type(2))) float v2f;
typedef __attribute__((ext_vector_type(8))) float v8f;

#define PI_F 3.14159265358979323846f

// ---------------------------------------------------------------------------
// Composite frequency-keep mapping: final 32-grid index -> layer-N grid index.
// Kept set at grid N is {0..15} U {N-16..N-1}; composing three 0.5-crops gives
//   u_N = u3 < 16 ? u3 : u3 + (N - 32)
// ---------------------------------------------------------------------------

// --------- Kernel A: row DFTs, 256 samples -> 32 kept frequency bins --------
// One wave per (b,c,h) row. Twiddle table in LDS, no transcendentals in loop.
__global__ __launch_bounds__(256) void k_row_dft(const float* __restrict__ X,
                                                 float2* __restrict__ R) {
  __shared__ float rows[8][256];
  __shared__ float2 tw[256];
  const int tid = threadIdx.x;
  {
    float s, c;
    __sincosf(-2.0f * PI_F * (float)tid / 256.0f, &s, &c);
    tw[tid] = make_float2(c, s);
  }
  const int wave = tid >> 5, lane = tid & 31;
  const int rowId = blockIdx.x * 8 + wave;  // 48*256 = 12288 rows, grid 1536
  const float* src = X + (size_t)rowId * 256;
  for (int j = lane; j < 256; j += 32) rows[wave][j] = src[j];
  __syncthreads();
  const int v3 = lane;
  const int v0 = (v3 < 16) ? v3 : v3 + 224;
  float re = 0.f, im = 0.f;
  for (int w = 0; w < 256; ++w) {
    const float2 T = tw[(v0 * w) & 255];
    const float x = rows[wave][w];
    re = fmaf(x, T.x, re);
    im = fmaf(x, T.y, im);
  }
  R[(size_t)rowId * 32 + v3] = make_float2(re, im);
}

// --------- Kernel B: column DFTs over kept columns, compact to 32x32 --------
// Writes layer-0 input planar layout X0[loc][b][c(4, padded)] with c=3 zeroed.
__global__ __launch_bounds__(256) void k_col_dft(const float2* __restrict__ R,
                                                 float* __restrict__ X0r,
                                                 float* __restrict__ X0i) {
  __shared__ float2 col[8][256];
  __shared__ float2 tw[256];
  const int tid = threadIdx.x;
  {
    float s, c;
    __sincosf(-2.0f * PI_F * (float)tid / 256.0f, &s, &c);
    tw[tid] = make_float2(c, s);
  }
  const int wave = tid >> 5, lane = tid & 31;
  const int pair = blockIdx.x * 8 + wave;   // 48*32 = 1536 pairs, grid 192
  const int bc = pair >> 5;                 // (b,c) plane
  const int v3 = pair & 31;
  const float2* src = R + (size_t)bc * 256 * 32 + v3;
  for (int j = lane; j < 256; j += 32) col[wave][j] = src[(size_t)j * 32];
  __syncthreads();
  const int u3 = lane;
  const int u0 = (u3 < 16) ? u3 : u3 + 224;
  float re = 0.f, im = 0.f;
  for (int h = 0; h < 256; ++h) {
    const float2 T = tw[(u0 * h) & 255];
    const float2 cv = col[wave][h];
    re += cv.x * T.x - cv.y * T.y;
    im += cv.x * T.y + cv.y * T.x;
  }
  const int loc = u3 * 32 + v3;
  const int b = bc / 3, cc = bc % 3;
  X0r[loc * 64 + b * 4 + cc] = re;
  X0i[loc * 64 + b * 4 + cc] = im;
  if (cc == 0) {  // zero the padded 4th input channel
    X0r[loc * 64 + b * 4 + 3] = 0.f;
    X0i[loc * 64 + b * 4 + 3] = 0.f;
  }
}

// --------- Conv layer: per-location complex GEMM via V_WMMA_F32_16X16X4_F32
// One block per kept frequency location. Phase 1: build Wf(u,v) = DFT of the
// 4x4 spatial kernel (16 taps, twiddles in LDS) while X is staged into LDS by
// GLOBAL_LOAD_ASYNC_TO_LDS_B128 (ASYNCcnt path). Phase 2: 8 waves compute the
// 16x64 complex output: wave = {Re,Im} x 4 N-tiles; Re = Xr*Wr - Xi*Wi,
// Im = Xr*Wi + Xi*Wr, each as f32 WMMA with k-loop over CIN/4.
template <int CIN>
__global__ __launch_bounds__(256) void k_fft_conv(const float* __restrict__ Xr_g,
                                                  const float* __restrict__ Xi_g,
                                                  const float* __restrict__ w_g,
                                                  int cin0, int gridN,
                                                  float* __restrict__ Yr_g,
                                                  float* __restrict__ Yi_g) {
  __shared__ __align__(16) float sXr[16 * CIN];
  __shared__ __align__(16) float sXi[16 * CIN];
  __shared__ __align__(16) float sWr[CIN * 64];
  __shared__ __align__(16) float sWi[CIN * 64];
  __shared__ float2 sT[16];

  const int tid = threadIdx.x;
  const int loc = blockIdx.x;            // 1024 kept locations
  const int u3 = loc >> 5, v3 = loc & 31;

  // ---- async stage X (planar [loc][16][CIN]) directly into LDS -------------
  // VDST = wave-relative LDS byte offset (low 32 bits of generic LDS pointer),
  // VADDR = 64-bit global address, no SADDR. Tracked by ASYNCcnt.
  {
    const float* gXr = Xr_g + (size_t)loc * 16 * CIN;
    const float* gXi = Xi_g + (size_t)loc * 16 * CIN;
    for (int e = tid * 4; e < 16 * CIN; e += 256 * 4) {
      unsigned lr = (unsigned)(size_t)&sXr[e];
      unsigned li = (unsigned)(size_t)&sXi[e];
      unsigned long long ar = (unsigned long long)(size_t)(gXr + e);
      unsigned long long ai = (unsigned long long)(size_t)(gXi + e);
      asm volatile("global_load_async_to_lds_b128 %0, %1, off"
                   :: "v"(lr), "v"(ar) : "memory");
      asm volatile("global_load_async_to_lds_b128 %0, %1, off"
                   :: "v"(li), "v"(ai) : "memory");
    }
  }

  if (tid < 16) {
    const int p = tid >> 2, q = tid & 3;
    const int u = (u3 < 16) ? u3 : u3 + (gridN - 32);
    const int v = (v3 < 16) ? v3 : v3 + (gridN - 32);
    const int ph = (u * p + v * q) % gridN;
    float s, c;
    __sincosf(-2.0f * PI_F * (float)ph / (float)gridN, &s, &c);
    sT[tid] = make_float2(c, s);
  }
  __syncthreads();

  // Wf[i][o] (LDS layout k-major) from 16 real taps
  for (int e = tid; e < CIN * 64; e += 256) {
    const int i = e >> 6, o = e & 63;
    float re = 0.f, im = 0.f;
    if (i < cin0) {
      const float* wp = w_g + (size_t)(o * cin0 + i) * 16;
#pragma unroll
      for (int t = 0; t < 16; ++t) {
        const float wv = wp[t];
        re = fmaf(wv, sT[t].x, re);
        im = fmaf(wv, sT[t].y, im);
      }
    }
    sWr[e] = re;
    sWi[e] = im;
  }
  asm volatile("s_wait_asynccnt 0" ::: "memory");  // my wave's X copies landed
  __syncthreads();                                 // publish to all waves

  const int wave = tid >> 5, lane = tid & 31;
  const int part = wave >> 2;             // 0 = Re tile, 1 = Im tile
  const int ntile = wave & 3;
  const int half = lane >> 4;             // K sub-pair select (ISA A/B layout)
  const int m = lane & 15;                // A row
  const int n = ntile * 16 + (lane & 15); // B/D column
  const float sgn = part ? 1.0f : -1.0f;  // Re tile subtracts Xi*Wi

  v8f acc = {};
#pragma unroll
  for (int kk = 0; kk < CIN / 4; ++kk) {
    const int klo = kk * 4 + 2 * half;
    v2f a1 = {sXr[m * CIN + klo], sXr[m * CIN + klo + 1]};
    v2f a2 = {sXi[m * CIN + klo], sXi[m * CIN + klo + 1]};
    v2f br = {sWr[klo * 64 + n], sWr[(klo + 1) * 64 + n]};
    v2f bi = {sWi[klo * 64 + n], sWi[(klo + 1) * 64 + n]};
    v2f b1 = part ? bi : br;              // per-lane select: EXEC stays full
    v2f b2 = part ? br : bi;
    b2 *= sgn;                            // f32 WMMA has no A/B NEG: do in VALU
    acc = __builtin_amdgcn_wmma_f32_16x16x4_f32(false, a1, false, b1,
                                                (short)0, acc, false, false);
    acc = __builtin_amdgcn_wmma_f32_16x16x4_f32(false, a2, false, b2,
                                                (short)0, acc, false, false);
  }

  float* outp = part ? Yi_g : Yr_g;       // CReLU is per-part relu
#pragma unroll
  for (int r = 0; r < 8; ++r) {
    const float vv = fmaxf(acc[r], 0.0f);
    outp[(size_t)loc * 1024 + (size_t)(r + 8 * half) * 64 + n] = vv;
  }
}

// --------- |.| + flatten to reference emb order: emb[b][o*1024 + loc] -------
__global__ __launch_bounds__(256) void k_abs_flatten(const float* __restrict__ Yr,
                                                     const float* __restrict__ Yi,
                                                     float* __restrict__ emb) {
  const int idx = blockIdx.x * 256 + threadIdx.x;  // 1,048,576 total
  const float re = Yr[idx], im = Yi[idx];
  const int loc = idx >> 10;
  const int rem = idx & 1023;
  const int b = rem >> 6, o = rem & 63;
  emb[(size_t)b * 65536 + o * 1024 + loc] = sqrtf(re * re + im * im);
}

// --------- MLP layer 0: (16 x 65536) @ (65536 x 64) + bias, relu ------------
__global__ __launch_bounds__(256) void k_mlp0(const float* __restrict__ emb,
                                              const float* __restrict__ mw0,
                                              const float* __restrict__ mb0,
                                              float* __restrict__ h0) {
  __shared__ float red[16 * 256];
  const int n = blockIdx.x, t = threadIdx.x;
  float acc[16];
#pragma unroll
  for (int b = 0; b < 16; ++b) acc[b] = 0.f;
  const float* wrow = mw0 + (size_t)n * 65536;
  for (int k = t; k < 65536; k += 256) {
    __builtin_prefetch(wrow + k + 2048, 0, 1);     // global_prefetch_b8
    const float wv = wrow[k];
#pragma unroll
    for (int b = 0; b < 16; ++b) acc[b] = fmaf(emb[(size_t)b * 65536 + k], wv, acc[b]);
  }
#pragma unroll
  for (int b = 0; b < 16; ++b) red[b * 256 + t] = acc[b];
  __syncthreads();
  for (int off = 128; off > 0; off >>= 1) {
    if (t < off) {
#pragma unroll
      for (int b = 0; b < 16; ++b) red[b * 256 + t] += red[b * 256 + t + off];
    }
    __syncthreads();
  }
  if (t < 16) h0[t * 64 + n] = fmaxf(red[t * 256] + mb0[n], 0.0f);
}

// --------- MLP tail: 64->32 relu, 32->2 ------------------------------------
__global__ __launch_bounds__(512) void k_mlp_tail(const float* __restrict__ h0,
                                                  const float* __restrict__ mw1,
                                                  const float* __restrict__ mb1,
                                                  const float* __restrict__ mw2,
                                                  const float* __restrict__ mb2,
                                                  float* __restrict__ out) {
  __shared__ float sh0[1024];
  __shared__ float sh1[512];
  const int t = threadIdx.x;
  sh0[t] = h0[t];
  sh0[t + 512] = h0[t + 512];
  __syncthreads();
  const int b = t >> 5, nn = t & 31;
  float a = mb1[nn];
  for (int k = 0; k < 64; ++k) a = fmaf(sh0[b * 64 + k], mw1[nn * 64 + k], a);
  sh1[b * 32 + nn] = fmaxf(a, 0.0f);
  __syncthreads();
  if (t < 32) {
    const int bb = t >> 1, j = t & 1;
    float a2 = mb2[j];
    for (int k = 0; k < 32; ++k) a2 = fmaf(sh1[bb * 32 + k], mw2[j * 32 + k], a2);
    out[bb * 2 + j] = a2;
  }
}

// ---------------------------------------------------------------------------
extern "C" void kernel_launch(void* const* d_in, const int* in_sizes, int n_in,
                              void* d_out, int out_size, void* d_ws, size_t ws_size,
                              hipStream_t stream) {
  const float* X_t = (const float*)d_in[0];   // (16,3,256,256)
  const float* w0  = (const float*)d_in[1];   // (64,3,4,4)
  const float* w1  = (const float*)d_in[2];   // (64,64,4,4)
  const float* w2  = (const float*)d_in[3];   // (64,64,4,4)
  const float* mw0 = (const float*)d_in[4];   // (64,65536)
  const float* mb0 = (const float*)d_in[5];
  const float* mw1 = (const float*)d_in[6];   // (32,64)
  const float* mb1 = (const float*)d_in[7];
  const float* mw2 = (const float*)d_in[8];   // (2,32)
  const float* mb2 = (const float*)d_in[9];
  float* out = (float*)d_out;                 // (16,2) f32

  float* W = (float*)d_ws;
  // workspace layout (floats; all offsets multiples of 4 -> 16B aligned)
  float2* R   = (float2*)(W + 0);             // 48*256*32 complex  = 786432 f
  float*  X0r = W + 786432;                   // 1024*16*4          =  65536 f
  float*  X0i = W + 851968;
  float*  Y0r = W + 917504;                   // 1024*16*64         = 1048576 f
  float*  Y0i = W + 1966080;
  float*  Y1r = W + 3014656;
  float*  Y1i = W + 4063232;
  float*  Y2r = W + 5111808;
  float*  Y2i = W + 6160384;
  float*  emb = W + 7208960;                  // 16*65536
  float*  h0  = W + 8257536;                  // 16*64
  // total 8,258,560 floats ~= 31.5 MB

  k_row_dft<<<1536, 256, 0, stream>>>(X_t, R);
  k_col_dft<<<192, 256, 0, stream>>>(R, X0r, X0i);
  k_fft_conv<4><<<1024, 256, 0, stream>>>(X0r, X0i, w0, 3, 256, Y0r, Y0i);
  k_fft_conv<64><<<1024, 256, 0, stream>>>(Y0r, Y0i, w1, 64, 128, Y1r, Y1i);
  k_fft_conv<64><<<1024, 256, 0, stream>>>(Y1r, Y1i, w2, 64, 64, Y2r, Y2i);
  k_abs_flatten<<<4096, 256, 0, stream>>>(Y2r, Y2i, emb);
  k_mlp0<<<64, 256, 0, stream>>>(emb, mw0, mb0, h0);
  k_mlp_tail<<<1, 512, 0, stream>>>(h0, mw1, mb1, mw2, mb2, out);
}